// DeepAir_27419071217940
// MI455X (gfx1250) — compile-verified
//
#include <hip/hip_runtime.h>
#include <hip/hip_bf16.h>

typedef __attribute__((ext_vector_type(2))) float v2f;
typedef __attribute__((ext_vector_type(8))) float v8f;

#define GAT_NT 256
#define N_NODES 300
#define N_EDGES 9000
#define BB 128
#define TT 24
#define GRU_H 12
#define OUT_F 1200

// ---------------------------------------------------------------------------
// Kernel 1: EdgeGAT (scalar-collapsed) + edge softmax + mean pool.
// One block per graph g in [0, B*T). Writes pooled scalar ps[g].
// pooled[g,o] = w_node[o]*ps[g] + gat_bias[o]   (folded into the GRU kernel)
// ---------------------------------------------------------------------------
__global__ __launch_bounds__(GAT_NT) void gat_pool_kernel(
    const float* __restrict__ x,      // [G, 300]
    const float* __restrict__ ew,     // [G, 9000]
    const int*   __restrict__ src,    // [9000]
    const int*   __restrict__ dst,    // [9000]
    const float* __restrict__ w_node, // [4]
    const float* __restrict__ w_edge, // [4]
    const float* __restrict__ attn_l, // [4]
    const float* __restrict__ attn_r, // [4]
    const float* __restrict__ attn_e, // [4]
    float* __restrict__ ps)           // [G]
{
    __shared__ float    xs[N_NODES];
    __shared__ unsigned menc[N_NODES];
    __shared__ float    den[N_NODES];
    __shared__ float    num[N_NODES];
    __shared__ float    ev[N_EDGES];
    __shared__ float    red[GAT_NT];

    const int g   = blockIdx.x;
    const int tid = threadIdx.x;
    const float* xg  = x  + (size_t)g * N_NODES;
    const float* ewg = ew + (size_t)g * N_EDGES;

    // scalar-collapsed attention coefficients
    float cl = 0.f, cr = 0.f, ce = 0.f;
#pragma unroll
    for (int o = 0; o < 4; ++o) {
        float wn = w_node[o];
        cl += wn * attn_l[o];
        cr += wn * attn_r[o];
        ce += w_edge[o] * attn_e[o];
    }

    for (int i = tid; i < N_NODES; i += GAT_NT) {
        xs[i]   = xg[i];
        menc[i] = 0u;      // < encoding of any finite float
        den[i]  = 0.f;
        num[i]  = 0.f;
    }
    __syncthreads();

    // Pass 1: edge logits (leaky_relu 0.2) + per-dst max (ordered-uint trick)
    for (int e = tid; e < N_EDGES; e += GAT_NT) {
        int s = src[e], d = dst[e];
        float w = __builtin_nontemporal_load(ewg + e);   // streamed once, NT hint
        float v = cl * xs[s] + cr * xs[d] + ce * w;
        v = v > 0.f ? v : 0.2f * v;
        ev[e] = v;
        unsigned u = __float_as_uint(v);
        unsigned en = (u & 0x80000000u) ? ~u : (u | 0x80000000u);
        atomicMax(&menc[d], en);
    }
    __syncthreads();

    // Pass 2: exp(e - max[dst]); segment sums of weight and weighted x[src]
    for (int e = tid; e < N_EDGES; e += GAT_NT) {
        int s = src[e], d = dst[e];
        unsigned u = menc[d];
        float mx = (u & 0x80000000u) ? __uint_as_float(u ^ 0x80000000u)
                                     : __uint_as_float(~u);
        float exv = __expf(ev[e] - mx);
        atomicAdd(&den[d], exv);
        atomicAdd(&num[d], exv * xs[s]);
    }
    __syncthreads();

    // Node reduce: mean over nodes of num/den (0 for isolated nodes)
    float acc = 0.f;
    for (int n = tid; n < N_NODES; n += GAT_NT) {
        float dn = den[n];
        if (dn > 0.f) acc += num[n] / dn;
    }
    red[tid] = acc;
    __syncthreads();
    for (int s2 = GAT_NT / 2; s2 > 0; s2 >>= 1) {
        if (tid < s2) red[tid] += red[tid + s2];
        __syncthreads();
    }
    if (tid == 0) ps[g] = red[0] * (1.0f / (float)N_NODES);
}

// ---------------------------------------------------------------------------
// Kernel 2: GRU (1 layer, input 4, hidden 12), one thread per batch row.
// Reconstructs pooled[b,t,:] = w_node*ps + gat_bias on the fly.
// ---------------------------------------------------------------------------
__global__ __launch_bounds__(BB) void gru_kernel(
    const float* __restrict__ ps,       // [B*T]
    const float* __restrict__ w_node,   // [4]
    const float* __restrict__ gat_bias, // [4]
    const float* __restrict__ w_ih,     // [36,4]
    const float* __restrict__ w_hh,     // [36,12]
    const float* __restrict__ b_ih,     // [36]
    const float* __restrict__ b_hh,     // [36]
    float* __restrict__ hT)             // [B,12]
{
    __shared__ float s_wih[36 * 4];
    __shared__ float s_whh[36 * 12];
    __shared__ float s_bih[36];
    __shared__ float s_bhh[36];
    __shared__ float s_wn[4];
    __shared__ float s_gb[4];

    const int tid = threadIdx.x;
    for (int i = tid; i < 144; i += BB) s_wih[i] = w_ih[i];
    for (int i = tid; i < 432; i += BB) s_whh[i] = w_hh[i];
    if (tid < 36) { s_bih[tid] = b_ih[tid]; s_bhh[tid] = b_hh[tid]; }
    if (tid < 4)  { s_wn[tid] = w_node[tid]; s_gb[tid] = gat_bias[tid]; }
    __syncthreads();

    const int b = tid;
    float h[GRU_H];
#pragma unroll
    for (int k = 0; k < GRU_H; ++k) h[k] = 0.f;

    for (int t = 0; t < TT; ++t) {
        float p = ps[b * TT + t];
        float xin[4];
#pragma unroll
        for (int o = 0; o < 4; ++o) xin[o] = s_wn[o] * p + s_gb[o];

        float gi[36], gh[36];
#pragma unroll
        for (int j = 0; j < 36; ++j) {
            float a = s_bih[j];
#pragma unroll
            for (int o = 0; o < 4; ++o) a += s_wih[j * 4 + o] * xin[o];
            gi[j] = a;
            float c = s_bhh[j];
#pragma unroll
            for (int k = 0; k < GRU_H; ++k) c += s_whh[j * GRU_H + k] * h[k];
            gh[j] = c;
        }
#pragma unroll
        for (int k = 0; k < GRU_H; ++k) {
            float r = 1.f / (1.f + __expf(-(gi[k] + gh[k])));
            float z = 1.f / (1.f + __expf(-(gi[12 + k] + gh[12 + k])));
            float n = tanhf(gi[24 + k] + r * gh[24 + k]);
            h[k] = (1.f - z) * n + z * h[k];
        }
    }
#pragma unroll
    for (int k = 0; k < GRU_H; ++k) hT[b * GRU_H + k] = h[k];
}

// ---------------------------------------------------------------------------
// Kernel 3: FC  out[128,1200] = hT[128,12] @ fc_w[1200,12]^T + fc_b
// via V_WMMA_F32_16X16X4_F32, one wave per 16x16 output tile, K=12 = 3x4.
// A layout (16x4 f32): lanes 0-15 -> K={0,1}, lanes 16-31 -> K={2,3}, M=lane&15.
// B layout (4x16 f32): N=lane&15; lanes 0-15 -> K={0,1}, lanes 16-31 -> K={2,3}.
// C/D (16x16 f32, 8 VGPRs): N=lane&15, M = v + 8*(lane>=16).
// ---------------------------------------------------------------------------
__global__ __launch_bounds__(256) void fc_wmma_kernel(
    const float* __restrict__ hT,    // [128,12]
    const float* __restrict__ fc_w,  // [1200,12]
    const float* __restrict__ fc_b,  // [1200]
    float* __restrict__ out)         // [128,1200]
{
    const int wave = threadIdx.x >> 5;
    const int lane = threadIdx.x & 31;
    const int tile = blockIdx.x * 8 + wave;  // 0..599
    const int mtile = tile & 7;              // 8 tiles of M=128
    const int ntile = tile >> 3;             // 75 tiles of N=1200
    const int half = lane >> 4;              // 0: K lo pair, 1: K hi pair
    const int idx  = lane & 15;
    const int row  = mtile * 16 + idx;       // M index for A
    const int col  = ntile * 16 + idx;       // N index for B / bias / store

    v8f c = {};
#pragma unroll
    for (int kb = 0; kb < 12; kb += 4) {
        const int k0 = kb + half * 2;
        v2f a, b;
        a.x = hT[row * GRU_H + k0];
        a.y = hT[row * GRU_H + k0 + 1];
        b.x = fc_w[col * GRU_H + k0];
        b.y = fc_w[col * GRU_H + k0 + 1];
        c = __builtin_amdgcn_wmma_f32_16x16x4_f32(
                false, a, false, b, (short)0, c, false, false);
    }

    const float bias  = fc_b[col];
    const int   mbase = mtile * 16 + half * 8;
#pragma unroll
    for (int v = 0; v < 8; ++v) {
        out[(mbase + v) * OUT_F + col] = c[v] + bias;
    }
}

// ---------------------------------------------------------------------------
extern "C" void kernel_launch(void* const* d_in, const int* in_sizes, int n_in,
                              void* d_out, int out_size, void* d_ws, size_t ws_size,
                              hipStream_t stream) {
    const float* x        = (const float*)d_in[0];
    const float* ew       = (const float*)d_in[1];
    const int*   src      = (const int*)d_in[2];
    const int*   dst      = (const int*)d_in[3];
    const float* w_node   = (const float*)d_in[4];
    const float* w_edge   = (const float*)d_in[5];
    const float* attn_l   = (const float*)d_in[6];
    const float* attn_r   = (const float*)d_in[7];
    const float* attn_e   = (const float*)d_in[8];
    const float* gat_bias = (const float*)d_in[9];
    const float* w_ih     = (const float*)d_in[10];
    const float* w_hh     = (const float*)d_in[11];
    const float* b_ih     = (const float*)d_in[12];
    const float* b_hh     = (const float*)d_in[13];
    const float* fc_w     = (const float*)d_in[14];
    const float* fc_b     = (const float*)d_in[15];
    float* out = (float*)d_out;

    float* ps = (float*)d_ws;        // [3072]
    float* hT = ps + BB * TT;        // [128*12]

    gat_pool_kernel<<<BB * TT, GAT_NT, 0, stream>>>(
        x, ew, src, dst, w_node, w_edge, attn_l, attn_r, attn_e, ps);
    gru_kernel<<<1, BB, 0, stream>>>(
        ps, w_node, gat_bias, w_ih, w_hh, b_ih, b_hh, hT);
    fc_wmma_kernel<<<75, 256, 0, stream>>>(hT, fc_w, fc_b, out);
}